// HGT_20323785244855
// MI455X (gfx1250) — compile-verified
//
#include <hip/hip_runtime.h>
#include <math.h>

// HGT forward for MI455X (gfx1250, wave32).
// Dense layers: V_WMMA_F32_16X16X4_F32, 64x16 output per wave (4 m-tiles reuse
// each B fragment), fully compile-time shapes (K and N) -> immediate-offset
// global loads, compile-time-specialized epilogues.
// Edge stage: VALU matvecs with LDS-staged 4KB relation matrices,
// segment softmax via ordered-uint atomicMax + f32 hardware atomic adds.

#define HID 128
#define NH  4
#define HD  32

typedef float v2f __attribute__((ext_vector_type(2)));
typedef float v8f __attribute__((ext_vector_type(8)));

__device__ __forceinline__ void atomAddF(float* p, float v) {
  __hip_atomic_fetch_add(p, v, __ATOMIC_RELAXED, __HIP_MEMORY_SCOPE_AGENT);
}

// order-preserving float<->uint encoding for atomicMax on floats
__device__ __forceinline__ unsigned encOrd(float f) {
  unsigned b = __float_as_uint(f);
  return (b & 0x80000000u) ? ~b : (b | 0x80000000u);
}
__device__ __forceinline__ float decOrd(unsigned u) {
  return (u & 0x80000000u) ? __uint_as_float(u & 0x7FFFFFFFu)
                           : __uint_as_float(~u);
}

__device__ __forceinline__ float gelu_f(float x) {
  return 0.5f * x * (1.0f + erff(x * 0.7071067811865475f));
}

// ---------------------------------------------------------------------------
// WMMA f32 GEMM:  C[M,ND] = post( pre(A)[M,KD] @ W[KD,ND] + b )
// Compile-time: KD, ND, PRE (0 none, 1 gelu, 2 affine), POST (0/1 relu),
// SK (gated skip mix with Hbuf). Wave computes 64 rows x 16 cols (MT=4 tiles
// sharing each B fragment -> 4 WMMAs per 2 B loads, all immediate offsets).
// blockDim = 128 (4 waves, consecutive n-tiles). Grid: (ceil(M/64), ND/64).
// M-tail rows are clamped to row M-1 (loads stay unconditional; stores guard).
// ---------------------------------------------------------------------------
template<int KD, int ND, int PRE, int POST, bool SK>
__global__ void gemm_wmma(const float* __restrict__ A, const float* __restrict__ W,
                          const float* __restrict__ bias, float* __restrict__ C,
                          int M,
                          const float* __restrict__ pscale,
                          const float* __restrict__ pshift,
                          const float* __restrict__ skipg,
                          const float* __restrict__ Hbuf)
{
  constexpr int MT = 4;
  const int wave = threadIdx.x >> 5;
  const int lane = threadIdx.x & 31;
  const int half = lane >> 4;          // 0: lanes 0-15, 1: lanes 16-31
  const int idx  = lane & 15;
  const int m0 = blockIdx.x * (16 * MT);
  const int n0 = ((blockIdx.y << 2) + wave) << 4;
  const int n = n0 + idx;

  const float* Arow[MT];
#pragma unroll
  for (int t = 0; t < MT; ++t) {
    int m = m0 + t * 16 + idx;
    if (m >= M) m = M - 1;             // clamp; surplus D rows discarded below
    Arow[t] = A + (size_t)m * KD;
  }
  const float* Wcol = W + n;           // + ka*ND becomes an immediate offset

  v8f acc[MT];
#pragma unroll
  for (int t = 0; t < MT; ++t) acc[t] = (v8f){0.f,0.f,0.f,0.f,0.f,0.f,0.f,0.f};

  const int kh = half << 1;            // lane's K sub-offset (0 or 2)
#pragma unroll
  for (int kk = 0; kk < KD; kk += 4) {
    v2f bv;                            // W[kk+kh][n], W[kk+kh+1][n]
    bv.x = Wcol[(size_t)(kk) * ND + (size_t)kh * ND];
    bv.y = Wcol[(size_t)(kk + 1) * ND + (size_t)kh * ND];
#pragma unroll
    for (int t = 0; t < MT; ++t) {
      v2f av = *(const v2f*)(Arow[t] + kk + kh);  // A[m][ka], A[m][ka+1]
      if (PRE == 1) { av.x = gelu_f(av.x); av.y = gelu_f(av.y); }
      else if (PRE == 2) {
        av.x = av.x * pscale[kk + kh]     + pshift[kk + kh];
        av.y = av.y * pscale[kk + kh + 1] + pshift[kk + kh + 1];
      }
      acc[t] = __builtin_amdgcn_wmma_f32_16x16x4_f32(
          false, av, false, bv, (short)0, acc[t], false, false);
    }
  }

  const float bn = bias[n];
  float sg = 0.f, og = 0.f;
  if (SK) {
    const float sv = 1.f / (1.f + expf(-skipg[0]));
    sg = sv; og = 1.f - sv;
  }

  // D element v of tile t -> row m0 + t*16 + v + 8*half, col n
  const int rbase = m0 + (half << 3);
  if (m0 + 16 * MT <= M) {             // fast path: full tile, no guards
    float* Cb = C + (size_t)rbase * ND + n;
    const float* Hb = SK ? (Hbuf + (size_t)rbase * ND + n) : nullptr;
#pragma unroll
    for (int t = 0; t < MT; ++t) {
#pragma unroll
      for (int v = 0; v < 8; ++v) {
        float val = acc[t][v] + bn;
        if (POST == 1) val = fmaxf(val, 0.f);
        if (SK) val = sg * val + og * Hb[(size_t)(t * 16 + v) * ND];
        Cb[(size_t)(t * 16 + v) * ND] = val;
      }
    }
  } else {
#pragma unroll
    for (int t = 0; t < MT; ++t) {
#pragma unroll
      for (int v = 0; v < 8; ++v) {
        const int mr = rbase + t * 16 + v;
        if (mr >= M) continue;
        float val = acc[t][v] + bn;
        if (POST == 1) val = fmaxf(val, 0.f);
        if (SK) val = sg * val + og * Hbuf[(size_t)mr * ND + n];
        C[(size_t)mr * ND + n] = val;
      }
    }
  }
}

// ---------------------------------------------------------------------------
// Edge stage kernels
// ---------------------------------------------------------------------------
__global__ void fill_u32(unsigned* __restrict__ p, unsigned v, int n) {
  int i = blockIdx.x * blockDim.x + threadIdx.x;
  if (i < n) p[i] = v;
}

// pass 1: per (edge, head) logit = (Q[dst,h] . (K[src,h] @ arel[h])) * pr[h]/sqrt(D)
__global__ void edge_logits(const int* __restrict__ src, const int* __restrict__ dst,
                            const float* __restrict__ Kt, const float* __restrict__ Qt,
                            const float* __restrict__ ar, const float* __restrict__ pr,
                            float* __restrict__ elog, unsigned* __restrict__ mbuf, int E)
{
  __shared__ float sar[NH * HD * HD];   // 16 KB
  for (int i = threadIdx.x; i < NH * HD * HD; i += blockDim.x) sar[i] = ar[i];
  __syncthreads();
  int t = blockIdx.x * blockDim.x + threadIdx.x;
  if (t >= E * NH) return;
  int e = t >> 2, h = t & 3;
  int s = src[e], d = dst[e];
  const float* Kr = Kt + (size_t)s * HID + h * HD;
  const float* Qr = Qt + (size_t)d * HID + h * HD;
  const float* Am = sar + h * HD * HD;
  float acc = 0.f;
  for (int f = 0; f < HD; ++f) {
    float kk = 0.f;
    for (int dd = 0; dd < HD; ++dd) kk += Kr[dd] * Am[dd * HD + f];
    acc += Qr[f] * kk;
  }
  float lg = acc * pr[h] * 0.17677669529663687f;   // 1/sqrt(32)
  elog[t] = lg;
  atomicMax(&mbuf[d * NH + h], encOrd(lg));
}

// pass 2: e = exp(logit - max); accumulate per-dst sum
__global__ void edge_expsum(const int* __restrict__ dst, float* __restrict__ elog,
                            const unsigned* __restrict__ mbuf,
                            float* __restrict__ sbuf, int E)
{
  int t = blockIdx.x * blockDim.x + threadIdx.x;
  if (t >= E * NH) return;
  int e = t >> 2, h = t & 3;
  int d = dst[e];
  float m  = decOrd(mbuf[d * NH + h]);
  float ev = expf(elog[t] - m);
  elog[t] = ev;
  atomAddF(&sbuf[d * NH + h], ev);
}

// pass 3: per (edge, h, f): v = V[src,h] @ mrel[h]; agg[dst] += v * a
__global__ void edge_apply(const int* __restrict__ src, const int* __restrict__ dst,
                           const float* __restrict__ Vt, const float* __restrict__ mr,
                           const float* __restrict__ elog, const float* __restrict__ sbuf,
                           float* __restrict__ agg, int E)
{
  __shared__ float smr[NH * HD * HD];   // 16 KB
  for (int i = threadIdx.x; i < NH * HD * HD; i += blockDim.x) smr[i] = mr[i];
  __syncthreads();
  long long t = (long long)blockIdx.x * blockDim.x + threadIdx.x;
  if (t >= (long long)E * HID) return;
  int e  = (int)(t >> 7);
  int hf = (int)(t & 127);
  int h = hf >> 5, f = hf & 31;
  int s = src[e], d = dst[e];
  float a = elog[e * NH + h] / fmaxf(sbuf[d * NH + h], 1e-16f);
  const float* Vr = Vt + (size_t)s * HID + h * HD;
  const float* Mr = smr + h * HD * HD + f;
  float acc = 0.f;
  for (int dd = 0; dd < HD; ++dd) acc += Vr[dd] * Mr[dd * HD];
  atomAddF(&agg[(size_t)d * HID + hf], acc * a);
}

// ---------------------------------------------------------------------------
// GraphNorm reductions (column = feature, over all M rows)
// ---------------------------------------------------------------------------
__global__ void colsum_k(const float* __restrict__ X, float* __restrict__ sum,
                         int M, int rpb) {
  int f = threadIdx.x;                    // blockDim = HID
  int r0 = blockIdx.x * rpb;
  int r1 = min(r0 + rpb, M);
  float acc = 0.f;
  for (int r = r0; r < r1; ++r) acc += X[(size_t)r * HID + f];
  atomAddF(&sum[f], acc);
}

__global__ void center_k(float* __restrict__ X, const float* __restrict__ sum1,
                         const float* __restrict__ gnms, float* __restrict__ csq,
                         int M, int rpb) {
  int f = threadIdx.x;
  float sub = (sum1[f] / (float)M) * gnms[f];
  int r0 = blockIdx.x * rpb;
  int r1 = min(r0 + rpb, M);
  float acc = 0.f;
  for (int r = r0; r < r1; ++r) {
    float c = X[(size_t)r * HID + f] - sub;
    X[(size_t)r * HID + f] = c;
    acc += c * c;
  }
  atomAddF(&csq[f], acc);
}

__global__ void scale_k(const float* __restrict__ csq, const float* __restrict__ gnw,
                        float* __restrict__ scl, int M) {
  int f = threadIdx.x;
  scl[f] = gnw[f] * (1.0f / sqrtf(csq[f] / (float)M + 1e-5f));
}

// ---------------------------------------------------------------------------
extern "C" void kernel_launch(void* const* d_in, const int* in_sizes, int n_in,
                              void* d_out, int out_size, void* d_ws, size_t ws_size,
                              hipStream_t stream)
{
  (void)n_in; (void)out_size; (void)ws_size;

  const float* x_p   = (const float*)d_in[0];
  const float* x_a   = (const float*)d_in[1];
  const int*   eiw   = (const int*)d_in[2];
  const int*   eir   = (const int*)d_in[3];
  const int*   eic   = (const int*)d_in[4];
  const float* Winp  = (const float*)d_in[5];
  const float* binp  = (const float*)d_in[6];
  const float* Wk_p  = (const float*)d_in[7];
  const float* Wq_p  = (const float*)d_in[8];
  const float* Wv_p  = (const float*)d_in[9];
  const float* Wa_p  = (const float*)d_in[10];
  const float* bk_p  = (const float*)d_in[11];
  const float* bq_p  = (const float*)d_in[12];
  const float* bv_p  = (const float*)d_in[13];
  const float* ba_p  = (const float*)d_in[14];
  const float* sk_p  = (const float*)d_in[15];
  const float* gnw_p = (const float*)d_in[16];
  const float* gnb_p = (const float*)d_in[17];
  const float* gnms_p= (const float*)d_in[18];
  const float* Wo_p  = (const float*)d_in[19];
  const float* bo_p  = (const float*)d_in[20];
  const float* Wina  = (const float*)d_in[21];
  const float* bina  = (const float*)d_in[22];
  const float* Wk_a  = (const float*)d_in[23];
  const float* Wq_a  = (const float*)d_in[24];
  const float* Wv_a  = (const float*)d_in[25];
  const float* Wa_a  = (const float*)d_in[26];
  const float* bk_a  = (const float*)d_in[27];
  const float* bq_a  = (const float*)d_in[28];
  const float* bv_a  = (const float*)d_in[29];
  const float* ba_a  = (const float*)d_in[30];
  const float* sk_a  = (const float*)d_in[31];
  const float* gnw_a = (const float*)d_in[32];
  const float* gnb_a = (const float*)d_in[33];
  const float* gnms_a= (const float*)d_in[34];
  const float* Wo_a  = (const float*)d_in[35];
  const float* bo_a  = (const float*)d_in[36];
  const float* ar_w  = (const float*)d_in[37];
  const float* mr_w  = (const float*)d_in[38];
  const float* pr_w  = (const float*)d_in[39];
  const float* ar_r  = (const float*)d_in[40];
  const float* mr_r  = (const float*)d_in[41];
  const float* pr_r  = (const float*)d_in[42];
  const float* ar_c  = (const float*)d_in[43];
  const float* mr_c  = (const float*)d_in[44];
  const float* pr_c  = (const float*)d_in[45];

  const int NP = in_sizes[0] / 64;
  const int NA = in_sizes[1] / 64;
  const int EW = in_sizes[2] / 2;
  const int ER = in_sizes[3] / 2;
  const int EC = in_sizes[4] / 2;
  int EMAX = EC > EW ? EC : EW; if (ER > EMAX) EMAX = ER;

  // ---- workspace carve-out ----
  float* ws = (float*)d_ws;
  size_t off = 0;
  auto alloc = [&](size_t nf) { float* p = ws + off; off += nf; return p; };
  float*    hp   = alloc((size_t)NP * HID);
  float*    ha   = alloc((size_t)NA * HID);
  float*    Kp   = alloc((size_t)NP * HID);
  float*    Qp   = alloc((size_t)NP * HID);
  float*    Vp   = alloc((size_t)NP * HID);
  float*    Ka   = alloc((size_t)NA * HID);
  float*    Qa   = alloc((size_t)NA * HID);
  float*    Va   = alloc((size_t)NA * HID);
  float*    aggp = alloc((size_t)NP * HID);
  float*    agga = alloc((size_t)NA * HID);
  float*    elog = alloc((size_t)EMAX * NH);
  unsigned* mbuf = (unsigned*)alloc((size_t)NP * NH);
  float*    sbuf = alloc((size_t)NP * NH);
  float*    sum1 = alloc(HID);
  float*    csq  = alloc(HID);
  float*    scl  = alloc(HID);
  // o buffers alias K buffers (K dead after edge stage)
  float* op = Kp;
  float* oa = Ka;

  auto grd = [](int M, int N) { return dim3((unsigned)((M + 63) / 64), (unsigned)(N / 64)); };
  const dim3 blk(128);

  // GEMM wrappers (template-specialized: K, N compile-time)
  auto gemm_in = [&](const float* A, const float* W, const float* b, float* C, int M) {
    gemm_wmma<64, HID, 0, 1, false><<<grd(M, HID), blk, 0, stream>>>(
        A, W, b, C, M, nullptr, nullptr, nullptr, nullptr);
  };
  auto gemm_kqv = [&](const float* A, const float* W, const float* b, float* C, int M) {
    gemm_wmma<HID, HID, 0, 0, false><<<grd(M, HID), blk, 0, stream>>>(
        A, W, b, C, M, nullptr, nullptr, nullptr, nullptr);
  };
  auto gemm_head = [&](const float* A, const float* W, const float* b, float* C, int M,
                       const float* sk, const float* hb) {
    gemm_wmma<HID, HID, 1, 0, true><<<grd(M, HID), blk, 0, stream>>>(
        A, W, b, C, M, nullptr, nullptr, sk, hb);
  };
  auto gemm_out = [&](const float* A, const float* W, const float* b, float* C, int M,
                      const float* ps, const float* pf) {
    gemm_wmma<HID, 64, 2, 0, false><<<grd(M, 64), blk, 0, stream>>>(
        A, W, b, C, M, ps, pf, nullptr, nullptr);
  };

  auto relation = [&](const int* ei, int E, const float* Kt, const float* Qt,
                      const float* Vt, const float* ar, const float* mr,
                      const float* pr, float* agg, int nd) {
    const int* sv = ei;
    const int* dv = ei + E;
    int nfill = nd * NH;
    fill_u32<<<(nfill + 255) / 256, 256, 0, stream>>>(mbuf, 0x007FFFFFu /*enc(-inf)*/, nfill);
    hipMemsetAsync(sbuf, 0, (size_t)nfill * sizeof(float), stream);
    int t1 = E * NH;
    edge_logits<<<(t1 + 255) / 256, 256, 0, stream>>>(sv, dv, Kt, Qt, ar, pr, elog, mbuf, E);
    edge_expsum<<<(t1 + 255) / 256, 256, 0, stream>>>(dv, elog, mbuf, sbuf, E);
    long long t3 = (long long)E * HID;
    edge_apply<<<(int)((t3 + 255) / 256), 256, 0, stream>>>(sv, dv, Vt, mr, elog, sbuf, agg, E);
  };

  auto head = [&](const float* agg, const float* hb, const float* Wa, const float* ba,
                  const float* sk, const float* gnw, const float* gnb, const float* gnms,
                  const float* Wo, const float* bo, float* obuf, float* outp, int M) {
    // o = sigmoid(s)*(gelu(agg)@Wa + ba) + (1-sigmoid(s))*h
    gemm_head(agg, Wa, ba, obuf, M, sk, hb);
    hipMemsetAsync(sum1, 0, HID * sizeof(float), stream);
    hipMemsetAsync(csq,  0, HID * sizeof(float), stream);
    const int rpb = 128;
    const int gb  = (M + rpb - 1) / rpb;
    colsum_k<<<gb, HID, 0, stream>>>(obuf, sum1, M, rpb);
    center_k<<<gb, HID, 0, stream>>>(obuf, sum1, gnms, csq, M, rpb);
    scale_k<<<1, HID, 0, stream>>>(csq, gnw, scl, M);
    // out = (c*scale + gnb) @ Wo + bo  (affine folded into A pre-load)
    gemm_out(obuf, Wo, bo, outp, M, scl, gnb);
  };

  // ---- pipeline ----
  hipMemsetAsync(aggp, 0, (size_t)NP * HID * sizeof(float), stream);
  hipMemsetAsync(agga, 0, (size_t)NA * HID * sizeof(float), stream);

  // input projections + relu
  gemm_in(x_p, Winp, binp, hp, NP);
  gemm_in(x_a, Wina, bina, ha, NA);
  // K/Q/V
  gemm_kqv(hp, Wk_p, bk_p, Kp, NP);
  gemm_kqv(hp, Wq_p, bq_p, Qp, NP);
  gemm_kqv(hp, Wv_p, bv_p, Vp, NP);
  gemm_kqv(ha, Wk_a, bk_a, Ka, NA);
  gemm_kqv(ha, Wq_a, bq_a, Qa, NA);
  gemm_kqv(ha, Wv_a, bv_a, Va, NA);

  // edge relations (writes + cites into agg_p; rev into agg_a)
  relation(eiw, EW, Ka, Qp, Va, ar_w, mr_w, pr_w, aggp, NP);
  relation(eic, EC, Kp, Qp, Vp, ar_c, mr_c, pr_c, aggp, NP);
  relation(eir, ER, Kp, Qa, Vp, ar_r, mr_r, pr_r, agga, NA);

  // heads -> outputs (tuple concatenated: out_p then out_a)
  float* out_p = (float*)d_out;
  float* out_a = out_p + (size_t)NP * 64;
  head(aggp, hp, Wa_p, ba_p, sk_p, gnw_p, gnb_p, gnms_p, Wo_p, bo_p, op, out_p, NP);
  head(agga, ha, Wa_a, ba_a, sk_a, gnw_a, gnb_a, gnms_a, Wo_a, bo_a, oa, out_a, NA);
}